// MultiTaskLoss_35287451304318
// MI455X (gfx1250) — compile-verified
//
#include <hip/hip_runtime.h>
#include <math.h>

typedef __attribute__((ext_vector_type(2))) float v2f;
typedef __attribute__((ext_vector_type(8))) float v8f;

// ---- constants from the reference ----
constexpr double R0 = 1.0 / 0.023, R1 = 1.0 / 0.003, R2 = 1.0 / 0.959, R3 = 1.0 / 0.015;
constexpr double RS = R0 + R1 + R2 + R3;
#define PW0 ((float)(R0 / RS * 4.0))
#define PW1 ((float)(R1 / RS * 4.0))
#define PW2 ((float)(R2 / RS * 4.0))
#define PW3 ((float)(R3 / RS * 4.0))
#define HUBER_DELTA 0.1f
#define EMA_ALPHA 0.85f

// Exact 32-lane sum using the matrix pipe: D = A x ones + 0.
// a0 = per-lane scalar, a1 = 0 -> every lane's value appears exactly once in A.
// With B == all-ones, D[m][n] = rowsum_m regardless of B's lane layout.
// Sum of a lane's 8 D VGPRs covers rows {0..7} (lanes 0-15) / {8..15} (lanes 16-31);
// one xor-16 shuffle-add merges the halves -> full wave sum in every lane.
__device__ __forceinline__ float wave_sum32(float s) {
  v2f a; a[0] = s;    a[1] = 0.0f;
  v2f b; b[0] = 1.0f; b[1] = 1.0f;
  v8f c = {};
  c = __builtin_amdgcn_wmma_f32_16x16x4_f32(false, a, false, b, (short)0, c, false, false);
  float r = c[0] + c[1] + c[2] + c[3] + c[4] + c[5] + c[6] + c[7];
  r += __shfl_xor(r, 16, 32);
  return r;
}

__global__ void zero_ws_kernel(float* ws) {
  if (threadIdx.x < 8) ws[threadIdx.x] = 0.0f;
}

// ws[0]=huber_sum ws[1]=sqi_sum ws[2]=w*nll sum ws[3]=w sum ws[4]=focal sum
__global__ void loss_main_kernel(const float* __restrict__ pred,
                                 const float4* __restrict__ plog4,
                                 const float* __restrict__ slog,
                                 const float* __restrict__ lab,
                                 const int* __restrict__ phl,
                                 const float* __restrict__ slab,
                                 const float* __restrict__ sqi,
                                 float* __restrict__ ws, int N) {
  const int tid = blockIdx.x * blockDim.x + threadIdx.x;
  const int stride = gridDim.x * blockDim.x;
  float hub = 0.0f, oks = 0.0f, wn = 0.0f, wsum = 0.0f, foc = 0.0f;
  for (int i = tid; i < N; i += stride) {
    __builtin_prefetch(pred + i + stride, 0, 1);
    __builtin_prefetch((const float*)(plog4 + i + stride), 0, 1);
    // ---- masked Huber (bis) ----
    float ok = (sqi[i] > 0.5f) ? 1.0f : 0.0f;
    float p = pred[i];
    float l = lab[i];
    float e = (p - l) * ok;                       // == p*ok - l*ok
    float ae = fabsf(e);
    hub += (ae < HUBER_DELTA) ? (0.5f * e * e)
                              : (HUBER_DELTA * (ae - 0.5f * HUBER_DELTA));
    oks += ok;
    // ---- weighted phase NLL (4-way log-softmax) ----
    float4 lg = plog4[i];
    float m = fmaxf(fmaxf(lg.x, lg.y), fmaxf(lg.z, lg.w));
    float se = __expf(lg.x - m) + __expf(lg.y - m) + __expf(lg.z - m) + __expf(lg.w - m);
    float lse = m + __logf(se);
    int lbl = phl[i] & 3;
    float ll = (lbl == 0) ? lg.x : (lbl == 1) ? lg.y : (lbl == 2) ? lg.z : lg.w;
    float w  = (lbl == 0) ? PW0  : (lbl == 1) ? PW1  : (lbl == 2) ? PW2  : PW3;
    wn += w * (lse - ll);
    wsum += w;
    // ---- focal loss (stim) ----
    float z = slog[i], y = slab[i];
    float lp  = (z >= 0.0f) ? (-log1pf(__expf(-z))) : (z - log1pf(__expf(z)));   // log_sigmoid(z)
    float l1  = (z <= 0.0f) ? (-log1pf(__expf(z)))  : (-z - log1pf(__expf(-z))); // log_sigmoid(-z)
    float bce = -(99.0f * y * lp + (1.0f - y) * l1);
    float pp  = 1.0f / (1.0f + __expf(-z));
    float pt  = pp * y + (1.0f - pp) * (1.0f - y);
    float at  = 0.99f * y + 0.01f * (1.0f - y);
    float om  = 1.0f - pt;
    foc += at * om * om * bce;
  }
  // wave-level reduction through the matrix pipe, then one atomic per wave
  float r0 = wave_sum32(hub);
  float r1 = wave_sum32(oks);
  float r2 = wave_sum32(wn);
  float r3 = wave_sum32(wsum);
  float r4 = wave_sum32(foc);
  if ((threadIdx.x & 31) == 0) {
    atomicAdd(&ws[0], r0);
    atomicAdd(&ws[1], r1);
    atomicAdd(&ws[2], r2);
    atomicAdd(&ws[3], r3);
    atomicAdd(&ws[4], r4);
  }
}

// One wave32 per batch row. EMA scan parallelized as a prefix composition of
// linear maps (m,c): ema = m*E_in + c, composed across lanes with shfl_up.
// ws[5]=step_loss sum  ws[6]=ema_loss sum  (both over B*(T-1) terms)
__global__ void mono_kernel(const float* __restrict__ pred,
                            const float* __restrict__ lab,
                            const int* __restrict__ phl,
                            float* __restrict__ ws, int B, int T) {
  const int lane = threadIdx.x & 31;
  const int wave = blockIdx.x * (blockDim.x >> 5) + (threadIdx.x >> 5);
  if (wave >= B) return;  // wave-uniform exit
  const long long base = (long long)wave * (long long)T;
  float E = 0.0f;                         // ema carry (value at end of prev chunk)
  float Xc = 0.0f, Lc = 0.0f; int Pc = 0; // prev-chunk lane-31 pred/label/phase
  float step_sum = 0.0f, ema_sum = 0.0f;
  const int nch = T >> 5;
  for (int c = 0; c < nch; ++c) {
    const int t = (c << 5) + lane;
    float x  = pred[base + t];
    float lb = lab[base + t];
    int   ph = phl[base + t];
    // per-position linear map; chunk0/lane0 is the constant init map (m=0,c=x0),
    // which makes all chunk-0 prefixes independent of E.
    float mm = EMA_ALPHA, cc = (1.0f - EMA_ALPHA) * x;
    if (c == 0 && lane == 0) { mm = 0.0f; cc = x; }
    // inclusive Hillis-Steele scan of map composition across the wave
    #pragma unroll
    for (int d = 1; d < 32; d <<= 1) {
      float pm = __shfl_up(mm, d, 32);
      float pc = __shfl_up(cc, d, 32);
      bool use = (lane >= d);
      float ncc = mm * pc + cc;
      float nmm = mm * pm;
      cc = use ? ncc : cc;
      mm = use ? nmm : mm;
    }
    float ema = mm * E + cc;              // ema[t]
    // loss index t' = c*32 - 1 + lane  (so t'+1 lives in the current chunk)
    float ema_p = __shfl_up(ema, 1, 32);
    float x_p   = __shfl_up(x,   1, 32);
    float lb_p  = __shfl_up(lb,  1, 32);
    int   ph_p  = __shfl_up(ph,  1, 32);
    ema_p = (lane == 0) ? E  : ema_p;
    x_p   = (lane == 0) ? Xc : x_p;
    lb_p  = (lane == 0) ? Lc : lb_p;
    ph_p  = (lane == 0) ? Pc : ph_p;
    float valid = (c > 0 || lane > 0) ? 1.0f : 0.0f;
    float dp = x - x_p;
    float dl = lb - lb_p;
    float de = ema - ema_p;
    float ind = (ph_p == 1) ? 1.0f : 0.0f;
    float rec = (ph_p == 3) ? 1.0f : 0.0f;
    float rp = (dp > 0.0f) ? dp : 0.0f;   // relu(d_pred)
    float rn = (dp < 0.0f) ? -dp : 0.0f;  // relu(-d_pred)
    step_sum += valid * (rp * ind * ((dl < 0.0f) ? 1.0f : 0.0f) +
                         rn * rec * ((dl > 0.0f) ? 1.0f : 0.0f));
    ema_sum  += valid * (rp * ind * ((de < 0.0f) ? 1.0f : 0.0f) +
                         rn * rec * ((de > 0.0f) ? 1.0f : 0.0f));
    // carries for next chunk
    E  = __shfl(ema, 31, 32);
    Xc = __shfl(x,   31, 32);
    Lc = __shfl(lb,  31, 32);
    Pc = __shfl(ph,  31, 32);
  }
  float s1 = wave_sum32(step_sum);
  float s2 = wave_sum32(ema_sum);
  if (lane == 0) {
    atomicAdd(&ws[5], s1);
    atomicAdd(&ws[6], s2);
  }
}

__global__ void finalize_kernel(const float* __restrict__ ws, float* __restrict__ out,
                                float invN, float invM) {
  if (threadIdx.x == 0 && blockIdx.x == 0) {
    float bis   = ws[0] / (ws[1] + 1e-6f);
    float phase = ws[2] / ws[3];
    float stim  = ws[4] * invN;
    float mono  = ws[5] * invM + 0.5f * (ws[6] * invM);
    float total = 1.0f * bis + 0.5f * phase + 0.3f * stim + 0.3f * mono;
    out[0] = total; out[1] = bis; out[2] = phase; out[3] = stim; out[4] = mono;
  }
}

extern "C" void kernel_launch(void* const* d_in, const int* in_sizes, int n_in,
                              void* d_out, int out_size, void* d_ws, size_t ws_size,
                              hipStream_t stream) {
  const float*  pred  = (const float*)d_in[0];   // (B,T,1) f32
  const float4* plog4 = (const float4*)d_in[1];  // (B,T,4) f32
  const float*  slog  = (const float*)d_in[2];   // (B,T)   f32
  const float*  lab   = (const float*)d_in[3];   // (B,T)   f32
  const int*    phl   = (const int*)d_in[4];     // (B,T)   int
  const float*  slab  = (const float*)d_in[5];   // (B,T)   f32
  const float*  sqi   = (const float*)d_in[6];   // (B,T)   f32
  float* out = (float*)d_out;
  float* ws  = (float*)d_ws;

  const int N = in_sizes[3];        // B*T
  const int T = 8192;
  const int B = N / T;

  zero_ws_kernel<<<1, 32, 0, stream>>>(ws);
  loss_main_kernel<<<1024, 256, 0, stream>>>(pred, plog4, slog, lab, phl, slab, sqi, ws, N);
  mono_kernel<<<(B + 7) / 8, 256, 0, stream>>>(pred, lab, phl, ws, B, T);
  finalize_kernel<<<1, 32, 0, stream>>>(ws, out, 1.0f / (float)N,
                                        1.0f / (float)(B * (T - 1)));
}